// cldice_54597624267058
// MI455X (gfx1250) — compile-verified
//
#include <hip/hip_runtime.h>
#include <hip/hip_bf16.h>
#include <stdint.h>

// ---------------- configuration ----------------
#define BLOCK 256
#define TILE  64
#define HALO  5                    // total stencil radius of soft_skel(ITER=3)
#define IW    (TILE + 2*HALO)      // 74
#define NPX   512
#define PPT   ((TILE*TILE)/BLOCK)  // 16 output pixels per thread

typedef __attribute__((ext_vector_type(2))) float v2f;
typedef __attribute__((ext_vector_type(8))) float v8f;

// ---------------- CDNA5 async global->LDS staging ----------------
#if __has_builtin(__builtin_amdgcn_global_load_async_to_lds_b32) && \
    __has_builtin(__builtin_amdgcn_s_wait_asynccnt)
#define CLD_ASYNC 1
typedef __attribute__((address_space(1))) int g_int;   // clang prints as '__device__ int *'
typedef __attribute__((address_space(3))) int l_int;   // clang prints as '__shared__ int *'
#else
#define CLD_ASYNC 0
#endif

__device__ __forceinline__ void async_copy_f32(const float* g, float* l) {
#if CLD_ASYNC
    __builtin_amdgcn_global_load_async_to_lds_b32(
        (g_int*)(uintptr_t)g,
        (l_int*)(uint32_t)(uintptr_t)l,   // low 32 bits of flat LDS addr = LDS offset
        0, 0);
#else
    *l = *g;
#endif
}

__device__ __forceinline__ void async_wait_all() {
#if CLD_ASYNC
    __builtin_amdgcn_s_wait_asynccnt(0);
#endif
}

// ---------------- wave32 sum via V_WMMA_F32_16X16X4_F32 ----------------
// A: vgpr0 = v, vgpr1 = 0  =>  A[m][0]=v_m, A[m][2]=v_{m+16}, others 0.
// B = all ones  =>  D[m][n] = v_m + v_{m+16}.
// Sum of the 8 D vgprs per lane covers rows 0..7 (lanes<16) / 8..15 (lanes>=16);
// adding the value from lane^16 gives the full 32-lane sum. Deterministic.
__device__ __forceinline__ float wave_sum32(float v) {
#if __has_builtin(__builtin_amdgcn_wmma_f32_16x16x4_f32)
    v2f a = {v, 0.0f};
    v2f b = {1.0f, 1.0f};
    v8f c = {};
    v8f d = __builtin_amdgcn_wmma_f32_16x16x4_f32(
        /*neg_a=*/false, a, /*neg_b=*/false, b,
        /*c_mod=*/(short)0, c, /*reuse_a=*/false, /*reuse_b=*/false);
    float s = ((d[0] + d[1]) + (d[2] + d[3])) + ((d[4] + d[5]) + (d[6] + d[7]));
    s += __shfl_xor(s, 16, 32);
    return s;
#else
    #pragma unroll
    for (int m = 16; m >= 1; m >>= 1) v += __shfl_xor(v, m, 32);
    return v;
#endif
}

// ---------------- fused soft_skel over one 64x64 tile ----------------
// erode  E(x) = min( 3x1-min , 1x3-min )   (OOB neighbors ignored: -inf pad of maxpool(-x))
// dilate D(x) = 3x3-max                    (OOB neighbors ignored)
// stage s (s=0..3): cur = E^s(img) valid at halo 5-s
//   nxt  = E(cur)                 valid at halo 4-s   (reused as next cur AND as dilate input)
//   open = D(nxt) at tile centers only
//   delta = relu(cur - open);  s==0: skel = delta; else skel += relu(delta - skel*delta)
template<bool CHECKED>
__device__ __forceinline__ void skel_pass(
    const float* __restrict__ img, const float* __restrict__ oth,
    int gr0, int gc0, float* bufA, float* bufB,
    float& acc_skel, float& acc_prod)
{
    const int tid = threadIdx.x;

    // ---- stage tile+halo into LDS (compact loop; async copies are fire-and-forget) ----
    #pragma unroll 2
    for (int i = tid; i < IW*IW; i += BLOCK) {
        int r = i / IW, c = i - r*IW;
        int gr = gr0 + r, gc = gc0 + c;
        if (CHECKED) {
            bool ok = ((unsigned)gr < (unsigned)NPX) && ((unsigned)gc < (unsigned)NPX);
            bufA[i] = ok ? img[gr*NPX + gc] : 0.0f;   // value never consumed when OOB
        } else {
            async_copy_f32(img + gr*NPX + gc, bufA + i);
        }
    }
    if (!CHECKED) async_wait_all();
    __syncthreads();

    float skel[PPT];
    float* cur = bufA;
    float* nxt = bufB;

    #pragma unroll
    for (int s = 0; s < 4; ++s) {
        const int g  = 4 - s;          // halo of E output
        const int a0 = HALO - g;       // array offset of E region
        const int RG = TILE + 2*g;

        // ---- nxt = E(cur) over (64+2g)^2 region (keep code compact) ----
        #pragma unroll 2
        for (int i = tid; i < RG*RG; i += BLOCK) {
            int r = i / RG, c = i - r*RG;
            int ar = a0 + r, ac = a0 + c;
            int off = ar*IW + ac;
            float ctr = cur[off];
            float vm = ctr, hm = ctr;
            if (CHECKED) {
                int gr = gr0 + ar, gc = gc0 + ac;
                if (gr >= 1)       vm = fminf(vm, cur[off - IW]);
                if (gr <= NPX - 2) vm = fminf(vm, cur[off + IW]);
                if (gc >= 1)       hm = fminf(hm, cur[off - 1]);
                if (gc <= NPX - 2) hm = fminf(hm, cur[off + 1]);
            } else {
                vm = fminf(vm, fminf(cur[off - IW], cur[off + IW]));
                hm = fminf(hm, fminf(cur[off - 1],  cur[off + 1]));
            }
            nxt[off] = fminf(vm, hm);
        }
        __syncthreads();

        // ---- delta + skeleton update at tile centers (skel[] must stay in VGPRs) ----
        #pragma unroll
        for (int k = 0; k < PPT; ++k) {
            int j = tid + k*BLOCK;                  // 0..4095 exactly covers 64x64
            int r = j >> 6, c = j & (TILE - 1);
            int off = (HALO + r)*IW + (HALO + c);
            float mx = nxt[off];
            if (CHECKED) {
                int gr = gr0 + HALO + r, gc = gc0 + HALO + c;  // always in-image
                bool up = gr >= 1, dn = gr <= NPX - 2;
                bool lf = gc >= 1, rt = gc <= NPX - 2;
                if (lf) mx = fmaxf(mx, nxt[off - 1]);
                if (rt) mx = fmaxf(mx, nxt[off + 1]);
                if (up) {
                    mx = fmaxf(mx, nxt[off - IW]);
                    if (lf) mx = fmaxf(mx, nxt[off - IW - 1]);
                    if (rt) mx = fmaxf(mx, nxt[off - IW + 1]);
                }
                if (dn) {
                    mx = fmaxf(mx, nxt[off + IW]);
                    if (lf) mx = fmaxf(mx, nxt[off + IW - 1]);
                    if (rt) mx = fmaxf(mx, nxt[off + IW + 1]);
                }
            } else {
                mx = fmaxf(mx, fmaxf(nxt[off - 1], nxt[off + 1]));
                mx = fmaxf(mx, fmaxf(nxt[off - IW], fmaxf(nxt[off - IW - 1], nxt[off - IW + 1])));
                mx = fmaxf(mx, fmaxf(nxt[off + IW], fmaxf(nxt[off + IW - 1], nxt[off + IW + 1])));
            }
            float delta = fmaxf(cur[off] - mx, 0.0f);
            if (s == 0) skel[k] = delta;
            else        skel[k] += fmaxf(delta - skel[k]*delta, 0.0f);
        }
        __syncthreads();

        float* t = cur; cur = nxt; nxt = t;
    }

    // ---- accumulate sum(skel) and sum(skel * other) ----
    #pragma unroll
    for (int k = 0; k < PPT; ++k) {
        int j = tid + k*BLOCK;
        int r = j >> 6, c = j & (TILE - 1);
        int gr = gr0 + HALO + r, gc = gc0 + HALO + c;
        float o = oth[gr*NPX + gc];
        acc_skel += skel[k];
        acc_prod += skel[k] * o;
    }
}

// ---------------- main kernel: one block per (batch, 64x64 tile) ----------------
__global__ void __launch_bounds__(BLOCK)
cldice_tile_kernel(const float* __restrict__ y_pred,
                   const float* __restrict__ y_true,
                   float* __restrict__ part)
{
    __shared__ float bufA[IW*IW];   // 21.9 KB
    __shared__ float bufB[IW*IW];   // 21.9 KB
    __shared__ float red[8][4];

    const int blk = blockIdx.x;             // 0..2047
    const int b  = blk >> 6;                // batch
    const int t  = blk & 63;
    const int ty = t >> 3, tx = t & 7;
    const int gr0 = ty*TILE - HALO, gc0 = tx*TILE - HALO;

    const float* p = y_pred + (size_t)b * NPX * NPX;
    const float* q = y_true + (size_t)b * NPX * NPX;

    const bool boundary = (tx == 0) | (tx == 7) | (ty == 0) | (ty == 7);

    // sums[0]=sum(skel_pred) sums[1]=sum(skel_pred*y_true)
    // sums[2]=sum(skel_true) sums[3]=sum(skel_true*y_pred)
    float sums[4] = {0.f, 0.f, 0.f, 0.f};
    #pragma unroll 1           // emit skel_pass body once per template variant
    for (int im = 0; im < 2; ++im) {
        const float* a = im ? q : p;
        const float* o = im ? p : q;
        float as = 0.f, ap = 0.f;
        if (boundary) skel_pass<true >(a, o, gr0, gc0, bufA, bufB, as, ap);
        else          skel_pass<false>(a, o, gr0, gc0, bufA, bufB, as, ap);
        if (im == 0) { sums[0] = as; sums[1] = ap; }
        else         { sums[2] = as; sums[3] = ap; }
    }

    // deterministic block reduction: WMMA wave-sum, then fixed-order cross-wave
    float r0 = wave_sum32(sums[0]), r1 = wave_sum32(sums[1]);
    float r2 = wave_sum32(sums[2]), r3 = wave_sum32(sums[3]);
    const int wave = threadIdx.x >> 5, lane = threadIdx.x & 31;
    if (lane == 0) { red[wave][0]=r0; red[wave][1]=r1; red[wave][2]=r2; red[wave][3]=r3; }
    __syncthreads();
    if (threadIdx.x == 0) {
        float a0=0.f, a1=0.f, a2=0.f, a3=0.f;
        #pragma unroll
        for (int w = 0; w < 8; ++w) { a0+=red[w][0]; a1+=red[w][1]; a2+=red[w][2]; a3+=red[w][3]; }
        part[blk*4+0]=a0; part[blk*4+1]=a1; part[blk*4+2]=a2; part[blk*4+3]=a3;
    }
}

// ---------------- finalize: reduce 2048 tile partials, emit clDice scalar ----------------
__global__ void __launch_bounds__(BLOCK)
cldice_final_kernel(const float* __restrict__ part, int ntiles, float* __restrict__ out)
{
    __shared__ float red[8][4];
    float s0=0.f, s1=0.f, s2=0.f, s3=0.f;
    #pragma unroll 1
    for (int t = threadIdx.x; t < ntiles; t += BLOCK) {   // 2048/256 = 8, uniform
        s0 += part[t*4+0]; s1 += part[t*4+1]; s2 += part[t*4+2]; s3 += part[t*4+3];
    }
    float r0 = wave_sum32(s0), r1 = wave_sum32(s1);
    float r2 = wave_sum32(s2), r3 = wave_sum32(s3);
    const int wave = threadIdx.x >> 5, lane = threadIdx.x & 31;
    if (lane == 0) { red[wave][0]=r0; red[wave][1]=r1; red[wave][2]=r2; red[wave][3]=r3; }
    __syncthreads();
    if (threadIdx.x == 0) {
        float Sp=0.f, Spt=0.f, St=0.f, Stp=0.f;
        #pragma unroll
        for (int w = 0; w < 8; ++w) { Sp+=red[w][0]; Spt+=red[w][1]; St+=red[w][2]; Stp+=red[w][3]; }
        const float SMOOTH = 1.0f;
        float tprec = (Spt + SMOOTH) / (Sp + SMOOTH);
        float tsens = (Stp + SMOOTH) / (St + SMOOTH);
        out[0] = 1.0f - 2.0f * (tprec * tsens) / (tprec + tsens);
    }
}

// ---------------- launch ----------------
extern "C" void kernel_launch(void* const* d_in, const int* in_sizes, int n_in,
                              void* d_out, int out_size, void* d_ws, size_t ws_size,
                              hipStream_t stream) {
    const float* y_pred = (const float*)d_in[0];   // (32,1,512,512) f32
    const float* y_true = (const float*)d_in[1];   // (32,1,512,512) f32
    float* out  = (float*)d_out;                   // 1 float
    float* part = (float*)d_ws;                    // 2048 * 4 floats = 32 KB scratch

    const int NTILES = 32 * 8 * 8;                 // batch * (512/64)^2
    cldice_tile_kernel<<<NTILES, BLOCK, 0, stream>>>(y_pred, y_true, part);
    cldice_final_kernel<<<1, BLOCK, 0, stream>>>(part, NTILES, out);
}